// GAT_72129680769630
// MI455X (gfx1250) — compile-verified
//
#include <hip/hip_runtime.h>
#include <hip/hip_bf16.h>
#include <stdint.h>

// GAT layer for MI455X (gfx1250, wave32).
// B=8, S=1024, DIM=128, HEADS=4, HEAD_DIM=32.
// 2-pass flash-style softmax over source dim; both GEMMs on
// V_WMMA_F32_16X16X4_F32 with LDS-staged B operands (conflict-free ds b64),
// batched global loads, rolled outer k-loops (bounded register pressure),
// and B-fragments hoisted per k-step so LDS latency overlaps the WMMA pipe.

typedef __attribute__((ext_vector_type(2))) float v2f;
typedef __attribute__((ext_vector_type(8))) float v8f;

#define GAT_B    8
#define GAT_S    1024
#define GAT_D    128
#define GAT_H    4
#define GAT_HD   32
#define FLT_BIG  3.402823466e+38f

// WMMA f32 16x16x4 layouts (ISA 7.12.2):
//   A 16x4 : lane l -> row M=l%16 ; VGPR v -> K = v + (l/16)*2
//   B 4x16 : lane l -> col N=l%16 ; VGPR v -> K = v + (l/16)*2
//   C 16x16: lane l -> col N=l%16 ; VGPR v -> row M = v + (l/16)*8

// ---------------------------------------------------------------------------
// Kernel 1: xp = x @ W^T + b_proj.
// Wave: 16 tokens x 128 cols, K=128. W staged in LDS (64KB) with XOR block
// swizzle: 4-dword block kk of row r is stored at block (kk ^ (r&15)) so
// half-wave lanes hit distinct banks -> conflict-free ds b64 reads.
// ---------------------------------------------------------------------------
__global__ void gat_proj_wmma(const float* __restrict__ x,
                              const float* __restrict__ W,
                              const float* __restrict__ bproj,
                              float* __restrict__ xp) {
  __shared__ float lW[GAT_D * GAT_D];              // 64KB exactly

  const int tid = threadIdx.x;                     // 0..127
  for (int idx = tid; idx < GAT_D * GAT_D / 4; idx += 128) {
    const int row = idx >> 5;
    const int c4  = idx & 31;
    const float4 w4 = *(const float4*)(W + (size_t)row * GAT_D + c4 * 4);
    const int c4s = c4 ^ (row & 15);               // XOR swizzle
    *(float4*)(lW + row * GAT_D + c4s * 4) = w4;
  }
  __syncthreads();

  const int lane    = tid & 31;
  const int wave    = tid >> 5;                    // 0..3
  const int tokBase = (blockIdx.x * 4 + wave) * 16;
  const int lm      = lane & 15;
  const int hs      = lane >> 4;

  const float* xrow = x + (size_t)(tokBase + lm) * GAT_D;

  v8f acc[8];
#pragma unroll
  for (int n = 0; n < 8; ++n) acc[n] = (v8f){0.f,0.f,0.f,0.f,0.f,0.f,0.f,0.f};

  // Keep this loop ROLLED: full unroll made ~64 fragments live -> spills.
#pragma clang loop unroll(disable)
  for (int g = 0; g < 8; ++g) {                    // 4 k-steps per group
    float4 xq[4];
#pragma unroll
    for (int t = 0; t < 4; ++t)                    // 4 batched b128 loads
      xq[t] = *(const float4*)(xrow + (g * 4 + t) * 4);
#pragma unroll
    for (int t = 0; t < 4; ++t) {
      const int kk = g * 4 + t;
      v2f a;                                       // lane-half picks K pair
      a.x = hs ? xq[t].z : xq[t].x;
      a.y = hs ? xq[t].w : xq[t].y;
      // Hoist all 8 B-fragments first: 4 paired ds b64 issue back-to-back,
      // waits retire with decreasing counts while WMMAs execute.
      v2f bf[8];
#pragma unroll
      for (int n = 0; n < 8; ++n) {
        const int row = n * 16 + lm;               // W row = output col
        const int blk = kk ^ (row & 15);           // undo swizzle
        bf[n] = *(const v2f*)(lW + row * GAT_D + blk * 4 + hs * 2);
      }
#pragma unroll
      for (int n = 0; n < 8; ++n)
        acc[n] = __builtin_amdgcn_wmma_f32_16x16x4_f32(
            false, a, false, bf[n], (short)0, acc[n], false, false);
    }
  }

#pragma unroll
  for (int n = 0; n < 8; ++n) {
    const float bv = bproj[n * 16 + lm];
#pragma unroll
    for (int v = 0; v < 8; ++v) {
      const int tok = tokBase + v + hs * 8;
      xp[(size_t)tok * GAT_D + n * 16 + lm] = acc[n][v] + bv;
    }
  }
}

// ---------------------------------------------------------------------------
// Kernel 2: a_src[t,h] = <xh[t,h,:], att_src[h,:]> ; same for a_dst.
// ---------------------------------------------------------------------------
__global__ void gat_attn_scores(const float* __restrict__ xp,
                                const float* __restrict__ att_src,
                                const float* __restrict__ att_dst,
                                float* __restrict__ a_src,
                                float* __restrict__ a_dst) {
  const int t = blockIdx.x * blockDim.x + threadIdx.x;
  if (t >= GAT_B * GAT_S * GAT_H) return;
  const int tok = t >> 2, h = t & 3;
  const float* row = xp + (size_t)tok * GAT_D + h * GAT_HD;
  float ss = 0.f, sd = 0.f;
#pragma unroll 8
  for (int d = 0; d < GAT_HD; ++d) {
    const float v = row[d];
    ss += v * att_src[h * GAT_HD + d];
    sd += v * att_dst[h * GAT_HD + d];
  }
  a_src[t] = ss;
  a_dst[t] = sd;
}

__device__ __forceinline__ float gat_logit(float as, float ad, int av, bool dead) {
  float v = as + ad;
  v = (av == 1) ? v : 0.f;          // zero non-edges BEFORE leaky (as in ref)
  v = (v >= 0.f) ? v : 0.2f * v;    // leaky_relu(0.2)
  return dead ? -FLT_BIG : v;
}

// ---------------------------------------------------------------------------
// Kernel 3: per-(b,j,h) softmax stats (max m, sum-exp s) over source dim i.
// Lanes = 32 consecutive j (coalesced A[b,i,j]); 8 waves split i; LDS merge.
// ---------------------------------------------------------------------------
__global__ void gat_softmax_stats(const int* __restrict__ Adj,
                                  const unsigned char* __restrict__ mask,
                                  const float* __restrict__ a_src,
                                  const float* __restrict__ a_dst,
                                  float* __restrict__ mOut,
                                  float* __restrict__ sOut) {
  __shared__ float lmx[8][32][GAT_H];
  __shared__ float lsm[8][32][GAT_H];

  const int lane = threadIdx.x & 31;
  const int wave = threadIdx.x >> 5;
  const int b    = blockIdx.x >> 5;
  const int jblk = blockIdx.x & 31;
  const int j    = jblk * 32 + lane;

  const bool dstMask = mask[b * GAT_S + j] != 0;
  float ad[GAT_H], m[GAT_H], s[GAT_H];
#pragma unroll
  for (int h = 0; h < GAT_H; ++h) {
    ad[h] = a_dst[((size_t)b * GAT_S + j) * GAT_H + h];
    m[h]  = -FLT_BIG;
    s[h]  = 0.f;
  }

  const int i0 = wave * 128;
#pragma clang loop unroll_count(4)
  for (int ii = 0; ii < 128; ++ii) {
    const int i  = i0 + ii;
    const int av = Adj[((size_t)b * GAT_S + i) * GAT_S + j];
    const bool dead = (mask[b * GAT_S + i] != 0) || dstMask;
    const float4 as4 = *(const float4*)(a_src + ((size_t)b * GAT_S + i) * GAT_H);
    const float asv[GAT_H] = {as4.x, as4.y, as4.z, as4.w};
#pragma unroll
    for (int h = 0; h < GAT_H; ++h) {
      const float v = gat_logit(asv[h], ad[h], av, dead);
      if (v <= m[h]) {
        s[h] += __expf(v - m[h]);          // one exp per element
      } else {
        s[h] = s[h] * __expf(m[h] - v) + 1.f;
        m[h] = v;
      }
    }
  }

#pragma unroll
  for (int h = 0; h < GAT_H; ++h) { lmx[wave][lane][h] = m[h]; lsm[wave][lane][h] = s[h]; }
  __syncthreads();

  if (threadIdx.x < 32 * GAT_H) {
    const int jl = threadIdx.x >> 2;
    const int h  = threadIdx.x & 3;
    float M = lmx[0][jl][h], S = lsm[0][jl][h];
#pragma unroll
    for (int w = 1; w < 8; ++w) {
      const float mw = lmx[w][jl][h], sw = lsm[w][jl][h];
      const float nm = fmaxf(M, mw);
      S = S * __expf(M - nm) + sw * __expf(mw - nm);
      M = nm;
    }
    const int jg = jblk * 32 + jl;
    mOut[((size_t)b * GAT_S + jg) * GAT_H + h] = M;
    sOut[((size_t)b * GAT_S + jg) * GAT_H + h] = S;
  }
}

// ---------------------------------------------------------------------------
// Kernel 4: out[b,j,h*32+d] = sum_i p(i,j,h)*xh[b,i,h,d] + xp + bias.
// Wave = (b,h,16-j tile); all 8 waves of a block share (b,h) -> xh chunk
// (128 i x 32 d) staged in LDS, K-pair interleaved:
//   (i,d) -> (i>>1)*96 + 2*d + (i&1)   (stride 96 => halves on disjoint banks)
// Each B-fragment = one conflict-free ds b64. Adjacency loads batched 8-deep
// per 4 K-steps; B-frags for the whole group hoisted so LDS latency overlaps
// the exp VALU chain; softmax weights built in-register in A-frag layout.
// ---------------------------------------------------------------------------
#define KC 128                                    // i-chunk length

__global__ void gat_aggregate_wmma(const int* __restrict__ Adj,
                                   const unsigned char* __restrict__ mask,
                                   const float* __restrict__ a_src,
                                   const float* __restrict__ a_dst,
                                   const float* __restrict__ mIn,
                                   const float* __restrict__ sIn,
                                   const float* __restrict__ xp,
                                   const float* __restrict__ bias,
                                   float* __restrict__ out) {
  __shared__ float lxh[(KC / 2) * 96];            // 24576B
  __shared__ float las[KC];
  __shared__ int   ldead[KC];

  const int tid  = threadIdx.x;
  const int lane = tid & 31;
  const int wave = tid >> 5;                      // 0..7
  const int gw   = blockIdx.x * 8 + wave;         // 0..2047
  const int jt   = gw & 63;
  const int bh   = gw >> 6;                       // same for whole block
  const int h    = bh & 3;
  const int b    = bh >> 2;
  const int lm   = lane & 15;
  const int hs   = lane >> 4;

  const int j = jt * 16 + lm;                     // A-frag row for this lane
  const bool  dstMask = mask[b * GAT_S + j] != 0;
  const float adv   = a_dst[((size_t)b * GAT_S + j) * GAT_H + h];
  const float mj    = mIn [((size_t)b * GAT_S + j) * GAT_H + h];
  const float inv_s = 1.0f / sIn[((size_t)b * GAT_S + j) * GAT_H + h];

  const int*   Acol   = Adj + (size_t)b * GAT_S * GAT_S + j;       // +i*S
  const float* xhbase = xp  + (size_t)b * GAT_S * GAT_D + h * GAT_HD;

  v8f acc0 = (v8f){0.f,0.f,0.f,0.f,0.f,0.f,0.f,0.f};
  v8f acc1 = (v8f){0.f,0.f,0.f,0.f,0.f,0.f,0.f,0.f};

#pragma clang loop unroll(disable)
  for (int c = 0; c < GAT_S / KC; ++c) {
    const int i0 = c * KC;
    __syncthreads();                              // LDS reuse fence
    // ---- cooperative staging ----
#pragma unroll
    for (int r = 0; r < (KC * GAT_HD) / 256; ++r) {
      const int elem = r * 256 + tid;             // 0..4095
      const int il = elem >> 5, d = elem & 31;    // coalesced 128B rows
      lxh[(il >> 1) * 96 + 2 * d + (il & 1)] =
          xhbase[(size_t)(i0 + il) * GAT_D + d];
    }
    if (tid < KC) {
      las[tid]   = a_src[((size_t)b * GAT_S + i0 + tid) * GAT_H + h];
      ldead[tid] = mask[b * GAT_S + i0 + tid];
    }
    __syncthreads();

    if (c + 1 < GAT_S / KC)                       // hint next adjacency chunk
      __builtin_prefetch(Acol + (size_t)(i0 + KC + (lane & 3)) * GAT_S, 0, 0);

    // ---- 32 K-steps, grouped by 4 so 8 adjacency loads are in flight.
    // Keep ROLLED to bound fragment live ranges (no scratch spills).
#pragma clang loop unroll(disable)
    for (int kq = 0; kq < KC / 16; ++kq) {
      int avv[8];
#pragma unroll
      for (int t = 0; t < 4; ++t) {
        const int li = kq * 16 + t * 4 + hs * 2;  // local i of A-frag K pair
        avv[2 * t]     = Acol[(size_t)(i0 + li)     * GAT_S];
        avv[2 * t + 1] = Acol[(size_t)(i0 + li + 1) * GAT_S];
      }
      // Hoist all B-fragments for this group: 8 ds b64 in flight at once.
      v2f bf0[4], bf1[4];
#pragma unroll
      for (int t = 0; t < 4; ++t) {
        const float* bp = lxh + ((kq * 16 + t * 4 + hs * 2) >> 1) * 96;
        bf0[t] = *(const v2f*)(bp + 2 * lm);          // d = 0..15
        bf1[t] = *(const v2f*)(bp + 2 * (lm + 16));   // d = 16..31
      }
#pragma unroll
      for (int t = 0; t < 4; ++t) {
        const int li = kq * 16 + t * 4 + hs * 2;  // even
        const bool d0 = (ldead[li]     != 0) || dstMask;
        const bool d1 = (ldead[li + 1] != 0) || dstMask;
        v2f a;
        a.x = __expf(gat_logit(las[li],     adv, avv[2*t],     d0) - mj) * inv_s;
        a.y = __expf(gat_logit(las[li + 1], adv, avv[2*t + 1], d1) - mj) * inv_s;

        acc0 = __builtin_amdgcn_wmma_f32_16x16x4_f32(
            false, a, false, bf0[t], (short)0, acc0, false, false);
        acc1 = __builtin_amdgcn_wmma_f32_16x16x4_f32(
            false, a, false, bf1[t], (short)0, acc1, false, false);
      }
    }
  }

  // Epilogue: C/D row M = v + hs*8, col = lm.  out = agg + xp + bias.
#pragma unroll
  for (int v = 0; v < 8; ++v) {
    const int jr = jt * 16 + v + hs * 8;
    const size_t base = ((size_t)b * GAT_S + jr) * GAT_D + h * GAT_HD + lm;
    out[base]      = acc0[v] + xp[base]      + bias[h * GAT_HD + lm];
    out[base + 16] = acc1[v] + xp[base + 16] + bias[h * GAT_HD + 16 + lm];
  }
}

// ---------------------------------------------------------------------------
extern "C" void kernel_launch(void* const* d_in, const int* in_sizes, int n_in,
                              void* d_out, int out_size, void* d_ws, size_t ws_size,
                              hipStream_t stream) {
  const float*         x       = (const float*)d_in[0];
  const int*           Adj     = (const int*)d_in[1];
  const unsigned char* mask    = (const unsigned char*)d_in[2]; // jax bool = 1B
  const float*         W       = (const float*)d_in[3];
  const float*         b_proj  = (const float*)d_in[4];
  const float*         att_src = (const float*)d_in[5];
  const float*         att_dst = (const float*)d_in[6];
  const float*         bias    = (const float*)d_in[7];
  float*               out     = (float*)d_out;

  // Workspace (floats): xp | a_src | a_dst | m | s   (~4.7 MB)
  float* ws    = (float*)d_ws;
  float* xp    = ws;
  float* a_src = xp    + (size_t)GAT_B * GAT_S * GAT_D;
  float* a_dst = a_src + (size_t)GAT_B * GAT_S * GAT_H;
  float* mbuf  = a_dst + (size_t)GAT_B * GAT_S * GAT_H;
  float* sbuf  = mbuf  + (size_t)GAT_B * GAT_S * GAT_H;

  gat_proj_wmma<<<128, 128, 0, stream>>>(x, W, b_proj, xp);
  gat_attn_scores<<<128, 256, 0, stream>>>(xp, att_src, att_dst, a_src, a_dst);
  gat_softmax_stats<<<256, 256, 0, stream>>>(Adj, mask, a_src, a_dst, mbuf, sbuf);
  gat_aggregate_wmma<<<256, 256, 0, stream>>>(Adj, mask, a_src, a_dst,
                                              mbuf, sbuf, xp, bias, out);
}